// Match_and_Mask_BD_and_BCE_loss_38053410242873
// MI455X (gfx1250) — compile-verified
//
#include <hip/hip_runtime.h>
#include <hip/hip_bf16.h>
#include <math.h>

#define BN 64
#define HN 512
#define WN 512
#define DN 256
#define TOTAL (BN * HN * WN)
#define TOTAL4 (TOTAL / 4)

typedef __attribute__((ext_vector_type(2))) float v2f;
typedef __attribute__((ext_vector_type(4))) float v4f;
typedef __attribute__((ext_vector_type(8))) float v8f;

// ---------------------------------------------------------------------------
// Kernel 0: zero the double accumulators in workspace (harness poisons ws).
// ---------------------------------------------------------------------------
__global__ void mm_init_acc(double* acc) {
    int i = threadIdx.x;
    if (i < 5) acc[i] = 0.0;
}

// ---------------------------------------------------------------------------
// Kernel 1: fused streaming pass, float4 (GLOBAL_LOAD_B128) per thread.
// Per 4 pixels: 1 b128 of x, 3 b128 of t (self/up/down rows), 2 scalar edge
// floats. Row re-reads are L0/L2 hits; DRAM traffic stays ~134 MB -> ~5.8us
// floor at 23.3 TB/s.
// acc[0]=sum BCE, acc[1]=sum(score*t), acc[2]=sum(score^2),
// acc[3]=S=nnz(t), acc[4]=C=boundary count (t==1 && 5-pt sum != 5).
// ---------------------------------------------------------------------------
__global__ __launch_bounds__(256) void mm_fused_reduce(
    const float* __restrict__ x, const float* __restrict__ t,
    double* __restrict__ acc) {
    const v4f* __restrict__ X4 = (const v4f*)x;
    const v4f* __restrict__ T4 = (const v4f*)t;
    const int stride = gridDim.x * blockDim.x;
    const v4f zero4 = {0.f, 0.f, 0.f, 0.f};

    float bce = 0.f, inter = 0.f, zz = 0.f, scnt = 0.f, ccnt = 0.f;

    for (int v = blockIdx.x * blockDim.x + threadIdx.x; v < TOTAL4; v += stride) {
        const int idx = v << 2;                 // first pixel of this float4
        const int w0  = idx & (WN - 1);         // column (float4 never crosses a row)
        const int h   = (idx >> 9) & (HN - 1);  // row within image (W = 512 = 2^9)

        v4f xv = X4[v];
        v4f tv = T4[v];
        v4f up = (h > 0)      ? T4[v - (WN / 4)] : zero4;
        v4f dn = (h < HN - 1) ? T4[v + (WN / 4)] : zero4;
        float lft = (w0 > 0)      ? t[idx - 1] : 0.f;
        float rgt = (w0 < WN - 4) ? t[idx + 4] : 0.f;

        float tl[6] = { lft, tv[0], tv[1], tv[2], tv[3], rgt };

        #pragma unroll
        for (int j = 0; j < 4; ++j) {
            float xe = xv[j];
            float te = tl[j + 1];

            // softplus(-x) = max(-x,0) + log1p(exp(-|x|))   (stable)
            float e  = __expf(-fabsf(xe));
            float sp = log1pf(e) + fmaxf(-xe, 0.f);
            // POS_WEIGHT=2: (1-t)*x + (1+t)*softplus(-x)
            bce += (1.f - te) * xe + (1.f + te) * sp;

            float sg = 1.f / (1.f + __expf(-xe));   // sigmoid
            inter += sg * te;
            zz    += sg * sg;

            if (te != 0.f) {                         // t in {0,1}
                scnt += 1.f;
                float ns = te + tl[j] + tl[j + 2] + up[j] + dn[j];
                if (ns != 5.f) ccnt += 1.f;          // Y = ns*t, Y==5 -> 0, count nnz
            }
        }
    }

    // wave32 reduction
    for (int o = 16; o > 0; o >>= 1) {
        bce   += __shfl_down(bce,   o, 32);
        inter += __shfl_down(inter, o, 32);
        zz    += __shfl_down(zz,    o, 32);
        scnt  += __shfl_down(scnt,  o, 32);
        ccnt  += __shfl_down(ccnt,  o, 32);
    }
    __shared__ float sm[8][5];
    int lane = threadIdx.x & 31;
    int wv   = threadIdx.x >> 5;                 // 8 waves per 256-thread block
    if (lane == 0) {
        sm[wv][0] = bce; sm[wv][1] = inter; sm[wv][2] = zz;
        sm[wv][3] = scnt; sm[wv][4] = ccnt;
    }
    __syncthreads();
    if (threadIdx.x < 5) {
        float s = 0.f;
        #pragma unroll
        for (int i = 0; i < 8; ++i) s += sm[i][threadIdx.x];
        atomicAdd(&acc[threadIdx.x], (double)s);  // global_atomic_add_f64
    }
}

// ---------------------------------------------------------------------------
// Kernel 2: gram-diagonals via V_WMMA_F32_16X16X4_F32 (codegen-confirmed).
// One wave per 16-row diagonal block; K-loop over 256 in steps of 4.
// A (16x4) and B (4x16) operands for a gram product share per-lane packing:
//   lane n (n = lane&15) holds {M[16d+n][k0+kh], M[16d+n][k0+kh+1]},
//   kh = 2*(lane>>4)   (ISA 32-bit 16x4 A layout: V0={K0,K2}, V1={K1,K3}).
// Diagonal extraction per C/D layout: m<8 -> vgpr m lane m; m>=8 -> vgpr m-8,
// lane m+16.
// ---------------------------------------------------------------------------
__global__ __launch_bounds__(128) void mm_cosine_wmma(
    const float* __restrict__ vf, const float* __restrict__ vb,
    const float* __restrict__ uf, float* __restrict__ diag) {
    const int lane = threadIdx.x & 31;
    const int wave = threadIdx.x >> 5;           // diag block 0..3
    const int row  = (wave << 4) + (lane & 15);  // matrix row for this lane
    const int kh   = (lane >> 4) << 1;           // 0 or 2

    v8f cfu = {}, cbu = {}, cff = {}, cbb = {}, cuu = {};

    for (int k0 = 0; k0 < DN; k0 += 4) {
        int base = row * DN + k0 + kh;
        v2f af = { vf[base], vf[base + 1] };
        v2f ab = { vb[base], vb[base + 1] };
        v2f au = { uf[base], uf[base + 1] };
        cfu = __builtin_amdgcn_wmma_f32_16x16x4_f32(false, af, false, au, (short)0, cfu, false, false);
        cbu = __builtin_amdgcn_wmma_f32_16x16x4_f32(false, ab, false, au, (short)0, cbu, false, false);
        cff = __builtin_amdgcn_wmma_f32_16x16x4_f32(false, af, false, af, (short)0, cff, false, false);
        cbb = __builtin_amdgcn_wmma_f32_16x16x4_f32(false, ab, false, ab, (short)0, cbb, false, false);
        cuu = __builtin_amdgcn_wmma_f32_16x16x4_f32(false, au, false, au, (short)0, cuu, false, false);
    }

    float dfu = 0.f, dbu = 0.f, dff = 0.f, dbb = 0.f, duu = 0.f;
    #pragma unroll
    for (int r = 0; r < 8; ++r) {
        if (lane == r || lane == r + 24) {
            dfu = cfu[r]; dbu = cbu[r]; dff = cff[r]; dbb = cbb[r]; duu = cuu[r];
        }
    }
    int m = (lane < 8) ? lane : (lane >= 24 ? lane - 16 : -1);
    if (m >= 0) {
        int b = (wave << 4) + m;
        diag[b]       = dfu;   // v_f . u_f
        diag[b + 64]  = dbu;   // v_b . u_f
        diag[b + 128] = dff;   // ||v_f||^2
        diag[b + 192] = dbb;   // ||v_b||^2
        diag[b + 256] = duu;   // ||u_f||^2
    }
}

// ---------------------------------------------------------------------------
// Kernel 3: combine everything into the final scalar.
// ---------------------------------------------------------------------------
__global__ __launch_bounds__(64) void mm_finalize(
    float* __restrict__ out, const double* __restrict__ acc,
    const float* __restrict__ diag) {
    __shared__ float sm[64];
    int b = threadIdx.x;

    float dfu = diag[b], dbu = diag[b + 64];
    float nf = fmaxf(sqrtf(diag[b + 128]), 1e-8f);
    float nb = fmaxf(sqrtf(diag[b + 192]), 1e-8f);
    float nu = fmaxf(sqrtf(diag[b + 256]), 1e-8f);
    float cf = dfu / (nf * nu);
    float cb = dbu / (nb * nu);
    float simf = fminf(fmaxf(cf, 1e-4f), 1.0f);
    float simb = fminf(fmaxf(cb, 1e-4f), 1.0f - 1e-4f);
    sm[b] = -(logf(simf) + 0.5f * log1pf(-simb));
    __syncthreads();
    for (int s = 32; s > 0; s >>= 1) {
        if (b < s) sm[b] += sm[b + s];
        __syncthreads();
    }
    if (b == 0) {
        double bce = acc[0] / (double)TOTAL;
        double I   = acc[1];
        double Z   = acc[2];
        double S   = acc[3];            // y_sum == S since t in {0,1}
        double C   = acc[4];
        double alpha = 2.0 * (1.0 - (C + 1e-5) / (S + 1e-5)) - 1.0;
        alpha = fmin(alpha, 0.8);
        double bd  = (Z + S - 2.0 * I + 1e-5) / (Z + S - (1.0 + alpha) * I + 1e-5);
        double mat = (double)sm[0] / 64.0;
        out[0] = (float)(0.5 * bce + 0.5 * bd + 0.1 * mat);
    }
}

// ---------------------------------------------------------------------------
extern "C" void kernel_launch(void* const* d_in, const int* in_sizes, int n_in,
                              void* d_out, int out_size, void* d_ws, size_t ws_size,
                              hipStream_t stream) {
    const float* x  = (const float*)d_in[0];   // low_res_logits (64,1,512,512)
    const float* t  = (const float*)d_in[1];   // target         (64,1,512,512)
    const float* vf = (const float*)d_in[2];   // v_f (64,256)
    const float* vb = (const float*)d_in[3];   // v_b (64,256)
    const float* uf = (const float*)d_in[4];   // u_f (64,256)

    double* acc = (double*)d_ws;                        // 5 doubles
    float*  dia = (float*)((char*)d_ws + 64);           // 5 x 64 floats

    mm_init_acc<<<1, 32, 0, stream>>>(acc);
    mm_fused_reduce<<<8192, 256, 0, stream>>>(x, t, acc);
    mm_cosine_wmma<<<1, 128, 0, stream>>>(vf, vb, uf, dia);
    mm_finalize<<<1, 64, 0, stream>>>((float*)d_out, acc, dia);
}